// GaussianBlurLayerNaive_14053132992834
// MI455X (gfx1250) — compile-verified
//
#include <hip/hip_runtime.h>
#include <hip/hip_bf16.h>
#include <math.h>

// ---------------------------------------------------------------------------
// GaussianBlurLayer for MI455X (gfx1250, wave32, WMMA).
//
// Z = B_sigma * X * B_sigma  per sample (B_sigma = symmetric 256x256 blur
// matrix capturing the Gaussian kernel + symmetric padding; only 32 distinct).
// Stage 0: build the 32 blur matrices (closed form, 8 MB in d_ws).
// Stage 1: Y = B * X   (column blur)  -> d_out
// Stage 2: Z = Y * B   (row blur)     -> d_out in place (block owns its rows)
// GEMMs use V_WMMA_F32_16X16X4_F32 with banded-K trimming (rad <= 80).
// ---------------------------------------------------------------------------

typedef __attribute__((ext_vector_type(2))) float v2f;
typedef __attribute__((ext_vector_type(8))) float v8f;
typedef int v4i __attribute__((vector_size(16)));  // matches builtin proto

#define IMG_N 256
#define RADMAX 80

#if defined(__has_builtin)
#if __has_builtin(__builtin_amdgcn_global_load_async_to_lds_b128) && \
    __has_builtin(__builtin_amdgcn_s_wait_asynccnt)
#define USE_ASYNC_LDS 1
#endif
#endif

static __device__ __forceinline__ v8f wmma_f32_4(v2f a, v2f b, v8f c) {
  // 8 args: (neg_a, A, neg_b, B, c_mod, C, reuse_a, reuse_b)
  return __builtin_amdgcn_wmma_f32_16x16x4_f32(
      false, a, false, b, (short)0, c, false, false);
}

// ---------------------------------------------------------------------------
// Build the 32 blur matrices. B[i][j] = w(j-i) + w(i+j+1) + w(2N-1-i-j),
// w(t) = exp(-0.5 (t/sigma)^2) / wsum for |t| <= rad, else 0.
// grid (32, 16), block 256: block (s, i-chunk); thread = column j.
// ---------------------------------------------------------------------------
__global__ void build_blurmat(const float* __restrict__ sigmas,
                              float* __restrict__ mats) {
  const int s  = blockIdx.x;
  const int i0 = blockIdx.y * 16;
  const int j  = threadIdx.x;

  const float sigma = sigmas[s];
  int rad = (int)floorf(4.0f * sigma + 0.5f);
  if (rad > RADMAX) rad = RADMAX;
  const float neg_half_inv_s2 = -0.5f / (sigma * sigma);

  float wsum = 1.0f;  // t = 0
  for (int t = 1; t <= rad; ++t)
    wsum += 2.0f * expf((float)(t * t) * neg_half_inv_s2);
  const float inv = 1.0f / wsum;

  float* M = mats + (size_t)s * IMG_N * IMG_N;
  for (int r = 0; r < 16; ++r) {
    const int i = i0 + r;
    float acc = 0.0f;
    int t = j - i;                       // direct
    if (t < 0) t = -t;
    if (t <= rad) acc += expf((float)(t * t) * neg_half_inv_s2);
    t = i + j + 1;                       // left reflection
    if (t <= rad) acc += expf((float)(t * t) * neg_half_inv_s2);
    t = 2 * IMG_N - 1 - i - j;           // right reflection
    if (t <= rad) acc += expf((float)(t * t) * neg_half_inv_s2);
    M[i * IMG_N + j] = acc * inv;
  }
}

// ---------------------------------------------------------------------------
// Per-sample GEMM stripe kernel.
// STAGE 0:  C = B_sigma * X      (A = blur matrix rows,  Bop = x sample)
// STAGE 1:  C = Y * B_sigma      (A = y sample rows,     Bop = blur matrix)
// Block: 256 threads (8 wave32), computes rows [i0, i0+32) x all 256 cols of
// one sample. A stripe staged in LDS (row stride 260 floats: conflict-free).
// Wave w owns col tiles {2w, 2w+1}; each wave has 2x2 accumulator tiles.
// ---------------------------------------------------------------------------
template <int STAGE>
__global__ void blur_gemm(const float* __restrict__ x,
                          const float* __restrict__ mats,
                          const float* __restrict__ sigmas,
                          const int* __restrict__ steps,
                          float* __restrict__ out) {
  constexpr int N = IMG_N;
  __shared__ float As[32][260];  // 33.3 KB, padded: zero-bank-conflict frags

  const int s   = blockIdx.y;
  const int i0  = blockIdx.x * 32;
  const int tid = threadIdx.x;

  const int   step  = steps[s];
  const float sigma = sigmas[step];
  int rad = (int)floorf(4.0f * sigma + 0.5f);
  if (rad > RADMAX) rad = RADMAX;

  const float* mat = mats + (size_t)step * N * N;
  float*       Cp  = out + (size_t)s * N * N;
  const float* Ap  = (STAGE == 0) ? (mat) : (const float*)Cp;
  const float* Bp  = (STAGE == 0) ? (x + (size_t)s * N * N) : mat;

  // ---- Stage-in A stripe (32 x 256) to LDS ----
#ifdef USE_ASYNC_LDS
  {
    // One async 16B chunk per thread: row = tid/8, 4 floats at col (tid%8)*32.
    const int r   = tid >> 3;
    const int col = (tid & 7) * 32;
    float* src = const_cast<float*>(Ap + (size_t)(i0 + r) * N + col);
    __builtin_amdgcn_global_load_async_to_lds_b128(
        (__attribute__((address_space(1))) v4i*)src,
        (__attribute__((address_space(3))) v4i*)&As[r][col],
        0, 0);
    __builtin_amdgcn_s_wait_asynccnt(0);
  }
#else
  {
    const int c4 = (tid & 63) * 4;
    for (int r = tid >> 6; r < 32; r += 4) {
      const float4 v = *(const float4*)(Ap + (size_t)(i0 + r) * N + c4);
      *(float4*)&As[r][c4] = v;
    }
  }
#endif
  __syncthreads();

  const int wave = tid >> 5;
  const int lane = tid & 31;
  const int m    = lane & 15;   // row (A) / col (B,C) within tile
  const int g    = lane >> 4;   // half-wave selects K pair / M half
  const int cb0  = wave * 32 + m;        // global column, tile ct=0
  const int kofs = 2 * g;                // this lane's K pair base

  v8f acc00 = {}, acc01 = {}, acc10 = {}, acc11 = {};

  // ---- Banded K range: blur matrix row/col i support is [i-rad, i+rad] ----
  int lo, hi;
  if (STAGE == 0) { lo = i0 - rad;        hi = i0 + 32 + rad; }        // A band
  else            { lo = wave * 32 - rad; hi = wave * 32 + 32 + rad; } // B band
  if (lo < 0) lo = 0;
  lo &= ~3;
  if (hi > N) hi = N;

  // Induction pointers: backend bumps addresses by immediates, no recompute.
  const float* a0p = &As[m][lo + kofs];
  const float* a1p = &As[16 + m][lo + kofs];
  const float* bq  = Bp + (size_t)(lo + kofs) * N + cb0;

#pragma unroll 2
  for (int k0 = lo; k0 < hi; k0 += 4) {
    // Pull upcoming B rows toward the WGP (speculative; safe past end).
    __builtin_prefetch((const void*)(bq + 8 * N), 0, 3);
    const v2f a0 = *(const v2f*)a0p;
    const v2f a1 = *(const v2f*)a1p;
    const v2f b0 = {bq[0], bq[N]};
    const v2f b1 = {bq[16], bq[N + 16]};
    a0p += 4;
    a1p += 4;
    bq += 4 * N;
    acc00 = wmma_f32_4(a0, b0, acc00);
    acc01 = wmma_f32_4(a0, b1, acc01);
    acc10 = wmma_f32_4(a1, b0, acc10);
    acc11 = wmma_f32_4(a1, b1, acc11);
  }

  // ---- Store: C/D layout => vgpr r holds (M = r + 8*g, N = m) ----
  const int colbase = wave * 32 + m;
  {
    float* cp = Cp + (size_t)(i0 + 0 * 16 + 8 * g) * N + colbase;
#pragma unroll
    for (int r = 0; r < 8; ++r) { cp[0] = acc00[r]; cp[16] = acc01[r]; cp += N; }
  }
  {
    float* cp = Cp + (size_t)(i0 + 1 * 16 + 8 * g) * N + colbase;
#pragma unroll
    for (int r = 0; r < 8; ++r) { cp[0] = acc10[r]; cp[16] = acc11[r]; cp += N; }
  }
}

// ---------------------------------------------------------------------------
extern "C" void kernel_launch(void* const* d_in, const int* in_sizes, int n_in,
                              void* d_out, int out_size, void* d_ws, size_t ws_size,
                              hipStream_t stream) {
  const float* x      = (const float*)d_in[0];  // (512, 256, 256) f32
  const float* sigmas = (const float*)d_in[1];  // (32,) f32
  const int*   steps  = (const int*)d_in[2];    // (512,) i32
  float*       out    = (float*)d_out;          // (512, 256, 256) f32
  float*       mats   = (float*)d_ws;           // 32 * 256 * 256 f32 = 8 MB

  // Build the 32 blur matrices (closed form incl. symmetric-pad reflections).
  build_blurmat<<<dim3(32, 16), 256, 0, stream>>>(sigmas, mats);

  // Y = B * X  (column blur), then Z = Y * B (row blur, in place per stripe).
  blur_gemm<0><<<dim3(8, 512), 256, 0, stream>>>(x, mats, sigmas, steps, out);
  blur_gemm<1><<<dim3(8, 512), 256, 0, stream>>>(x, mats, sigmas, steps, out);
}